// SinkhornDistance_37615323578834
// MI455X (gfx1250) — compile-verified
//
#include <hip/hip_runtime.h>
#include <math.h>

typedef __attribute__((ext_vector_type(2))) float v2f;
typedef __attribute__((ext_vector_type(8))) float v8f;

#define NB 4
#define P 1024
#define DD 16
#define EPS 0.1f
#define INV_EPS 10.0f
#define MAX_ITER 100
#define THRESH 0.1f

// workspace layout (float offsets)
#define WS_U     0       // [4][1024] dual u (updated in place)
#define WS_V     4096    // [4][1024] dual v (updated in place)
#define WS_LOGMU 8192
#define WS_LOGNU 12288
#define WS_ERRP  16384   // [2][256] double-buffered per-block error partials
#define WS_COSTP 16896   // [4][64] cost partials

__global__ void k_init(const float* __restrict__ wx, const float* __restrict__ wy,
                       float* __restrict__ ws) {
  int idx = blockIdx.x * blockDim.x + threadIdx.x; // 0..4095
  ws[WS_U + idx] = 0.f;
  ws[WS_V + idx] = 0.f;
  ws[WS_LOGMU + idx] = __logf(wx[idx] + 1e-8f);
  ws[WS_LOGNU + idx] = __logf(wy[idx] + 1e-8f);
  if (idx < 512) ws[WS_ERRP + idx] = 1e30f;  // "not converged" for both parities
}

// C[n][i][j] = sum_d |x[n][i][d] - y[n][j][d]|, plus transposed copy CT[n][j][i]
__global__ void k_cost(const float* __restrict__ x, const float* __restrict__ y,
                       float* __restrict__ C, float* __restrict__ CT) {
  __shared__ float xs[32 * DD];
  __shared__ float ys[32 * DD];
  __shared__ float tile[32][33];
  int n = blockIdx.z, bi = blockIdx.y, bj = blockIdx.x;
  int tid = threadIdx.x;
  const float* xb = x + ((size_t)n * P + (size_t)bi * 32) * DD;
  const float* yb = y + ((size_t)n * P + (size_t)bj * 32) * DD;
  xs[tid] = xb[tid]; xs[tid + 256] = xb[tid + 256];
  ys[tid] = yb[tid]; ys[tid + 256] = yb[tid + 256];
  __syncthreads();
  int jl = tid & 31;
  int il0 = (tid >> 5) * 4;
  for (int q = 0; q < 4; ++q) {
    int il = il0 + q;
    float s = 0.f;
#pragma unroll
    for (int d = 0; d < DD; ++d) s += fabsf(xs[il * DD + d] - ys[jl * DD + d]);
    C[((size_t)n * P + (size_t)bi * 32 + il) * P + (size_t)bj * 32 + jl] = s;
    tile[il][jl] = s;
  }
  __syncthreads();
  for (int q = 0; q < 4; ++q) {
    int jr = il0 + q;   // row of CT tile
    int ic = jl;        // col of CT tile (coalesced across lanes)
    CT[((size_t)n * P + (size_t)bj * 32 + jr) * P + (size_t)bi * 32 + ic] = tile[ic][jr];
  }
}

// out[n][r] = EPS*(logw[n][r] - logsumexp_j((dual[n][j] - Cm[n][r][j]) * INV_EPS))
// Block = (n, group of 16 rows), 256 threads = 8 waves; wave w owns a 128-col slice.
// Freeze flag derived per-block from previous iteration's error partials (fixed-tree
// reduction, identical in every block -> consistent decision, no extra kernel).
// Pass A: per-row max (wave shfl reduce). Pass B: exp-tile x ones via V_WMMA_F32_16X16X4_F32.
template <bool CERR>
__global__ void k_lse(const float* __restrict__ Cm, const float* __restrict__ dual,
                      const float* __restrict__ logw, float* out,
                      const float* uold,
                      const float* __restrict__ errPrev, float* __restrict__ errCur) {
  __shared__ float red[256];
  __shared__ float dlds[P];
  __shared__ float wpart[8 * 16];
  __shared__ float mrow[16];
  __shared__ float partial[8][16];
  __shared__ float errtmp[16];
  int tid = threadIdx.x;
  // ---- derive done(it-1) ----
  red[tid] = errPrev[tid];
  __syncthreads();
#pragma unroll
  for (int s = 128; s > 0; s >>= 1) {
    if (tid < s) red[tid] += red[tid + s];
    __syncthreads();
  }
  int done = __builtin_amdgcn_readfirstlane((red[0] * 0.25f < THRESH) ? 1 : 0);
  if (done) {
    // propagate freeze into this iteration's error buffer, then skip all work
    if (CERR && tid == 0) errCur[blockIdx.y * 64 + blockIdx.x] = 0.f;
    return;
  }
  int n = blockIdx.y;
  int i0 = blockIdx.x * 16;
  int lane = tid & 31, w = tid >> 5;
  int c0 = w * 128;
  for (int idx = tid; idx < P; idx += 256) dlds[idx] = dual[n * P + idx];
  __syncthreads();
  float du0 = dlds[c0 + lane], du1 = dlds[c0 + lane + 32];
  float du2 = dlds[c0 + lane + 64], du3 = dlds[c0 + lane + 96];
  const float* Cb = Cm + ((size_t)n * P + i0) * P;
  // ---- Pass A: row maxima over this wave's 128-column slice (unscaled; scale later) ----
  for (int r = 0; r < 16; ++r) {
    const float* row = Cb + (size_t)r * P + c0;
    float t0 = du0 - row[lane];
    float t1 = du1 - row[lane + 32];
    float t2 = du2 - row[lane + 64];
    float t3 = du3 - row[lane + 96];
    float tm = fmaxf(fmaxf(t0, t1), fmaxf(t2, t3));
#pragma unroll
    for (int off = 16; off > 0; off >>= 1) tm = fmaxf(tm, __shfl_xor(tm, off, 32));
    if (lane == 0) wpart[w * 16 + r] = tm;
  }
  __syncthreads();
  if (tid < 16) {
    float m = wpart[tid];
#pragma unroll
    for (int ww = 1; ww < 8; ++ww) m = fmaxf(m, wpart[ww * 16 + tid]);
    mrow[tid] = m * INV_EPS;  // INV_EPS > 0 so max commutes with scaling
  }
  __syncthreads();
  // ---- Pass B: sum_j exp(t - m) via fp32 WMMA (A = exp tile 16x4, B = ones 4x16) ----
  int M = lane & 15;              // A-matrix row for this lane
  int k0 = (lane >> 4) << 1;      // lanes 0-15 -> K=0,1 ; lanes 16-31 -> K=2,3
  const float* Crow = Cb + (size_t)M * P;
  float mr = mrow[M];
  v8f acc = {0.f, 0.f, 0.f, 0.f, 0.f, 0.f, 0.f, 0.f};
  v2f ones; ones.x = 1.f; ones.y = 1.f;
#pragma unroll 4
  for (int c = 0; c < 32; ++c) {
    int j = c0 + 4 * c + k0;
    float ta = (dlds[j]     - Crow[j]    ) * INV_EPS - mr;
    float tb = (dlds[j + 1] - Crow[j + 1]) * INV_EPS - mr;
    v2f a; a.x = __expf(ta); a.y = __expf(tb);
    acc = __builtin_amdgcn_wmma_f32_16x16x4_f32(false, a, false, ones, (short)0, acc,
                                                false, false);
  }
  // D layout: VGPR r, lane 0 -> rows 0..7 (col 0), lane 16 -> rows 8..15 (col 0)
  if (lane == 0 || lane == 16) {
    int rbase = (lane == 0) ? 0 : 8;
#pragma unroll
    for (int r = 0; r < 8; ++r) partial[w][rbase + r] = acc[r];
  }
  __syncthreads();
  if (tid < 16) {
    float S = 0.f;
#pragma unroll
    for (int ww = 0; ww < 8; ++ww) S += partial[ww][tid];
    float L = mrow[tid] + __logf(S);
    int gi = n * P + i0 + tid;
    float o = EPS * (logw[gi] - L);
    if (CERR) {
      float prev = uold[gi];     // read old u before in-place overwrite
      errtmp[tid] = fabsf(o - prev);
    }
    out[gi] = o;                 // in-place u update: block owns these 16 rows
  }
  if (CERR) {
    __syncthreads();
    if (tid == 0) {
      float e = 0.f;
#pragma unroll
      for (int r = 0; r < 16; ++r) e += errtmp[r];
      errCur[blockIdx.y * 64 + blockIdx.x] = e;  // deterministic per-block partial
    }
  }
}

// pi = exp((u_i + v_j - C_ij)/eps); per-block cost partials via fixed-tree reduce.
__global__ void k_pi(const float* __restrict__ Cmat, const float* __restrict__ ws,
                     float* __restrict__ pi, float* __restrict__ costP) {
  __shared__ float red[256];
  int n = blockIdx.y;
  int i0 = blockIdx.x * 16;
  int tid = threadIdx.x;
  const float* u = ws + WS_U + n * P;
  const float* v = ws + WS_V + n * P;
  const float* Cb = Cmat + (size_t)n * P * P + (size_t)i0 * P;
  float* pib = pi + (size_t)n * P * P + (size_t)i0 * P;
  float v0 = v[tid], v1 = v[tid + 256], v2 = v[tid + 512], v3 = v[tid + 768];
  float csum = 0.f;
  for (int r = 0; r < 16; ++r) {
    float ur = u[i0 + r];
    const float* crow = Cb + (size_t)r * P;
    float* prow = pib + (size_t)r * P;
    float c0 = crow[tid], c1 = crow[tid + 256], c2 = crow[tid + 512], c3 = crow[tid + 768];
    float p0 = __expf((ur + v0 - c0) * INV_EPS);
    float p1 = __expf((ur + v1 - c1) * INV_EPS);
    float p2 = __expf((ur + v2 - c2) * INV_EPS);
    float p3 = __expf((ur + v3 - c3) * INV_EPS);
    prow[tid] = p0; prow[tid + 256] = p1; prow[tid + 512] = p2; prow[tid + 768] = p3;
    csum += p0 * c0 + p1 * c1 + p2 * c2 + p3 * c3;
  }
  red[tid] = csum;
  __syncthreads();
#pragma unroll
  for (int s = 128; s > 0; s >>= 1) {
    if (tid < s) red[tid] += red[tid + s];
    __syncthreads();
  }
  if (tid == 0) costP[n * 64 + blockIdx.x] = red[0];
}

// Final deterministic fixed-order sum of 64 partials per batch.
__global__ void k_pifin(const float* __restrict__ costP, float* __restrict__ cost) {
  int t = threadIdx.x;
  if (t < NB) {
    float s = 0.f;
    for (int k = 0; k < 64; ++k) s += costP[t * 64 + k];
    cost[t] = s;
  }
}

extern "C" void kernel_launch(void* const* d_in, const int* in_sizes, int n_in,
                              void* d_out, int out_size, void* d_ws, size_t ws_size,
                              hipStream_t stream) {
  (void)in_sizes; (void)n_in; (void)out_size; (void)ws_size;
  const float* x  = (const float*)d_in[0];
  const float* y  = (const float*)d_in[1];
  const float* wx = (const float*)d_in[2];
  const float* wy = (const float*)d_in[3];
  float* out  = (float*)d_out;
  float* cost = out;                              // [4]
  float* pi   = out + 4;                          // [4,1024,1024]
  float* C    = out + 4 + (size_t)NB * P * P;     // [4,1024,1024]
  float* ws   = (float*)d_ws;
  float* CT   = pi;  // pi region doubles as transposed-C scratch during iterations

  k_init<<<16, 256, 0, stream>>>(wx, wy, ws);
  k_cost<<<dim3(32, 32, NB), 256, 0, stream>>>(x, y, C, CT);
  for (int it = 0; it < MAX_ITER; ++it) {
    float* errPrev = ws + WS_ERRP + ((it + 1) & 1) * 256;  // err(it-1)
    float* errCur  = ws + WS_ERRP + (it & 1) * 256;        // err(it)
    // u update (in place; each block owns its 16 rows)
    k_lse<true><<<dim3(64, NB), 256, 0, stream>>>(C, ws + WS_V, ws + WS_LOGMU,
                                                  ws + WS_U, ws + WS_U,
                                                  errPrev, errCur);
    // v update (freeze decision uses err(it-1), matching reference ordering)
    k_lse<false><<<dim3(64, NB), 256, 0, stream>>>(CT, ws + WS_U, ws + WS_LOGNU,
                                                   ws + WS_V, nullptr,
                                                   errPrev, nullptr);
  }
  k_pi<<<dim3(64, NB), 256, 0, stream>>>(C, ws, pi, ws + WS_COSTP);
  k_pifin<<<1, 32, 0, stream>>>(ws + WS_COSTP, cost);
}